// CapsuleLayer_30897994728312
// MI455X (gfx1250) — compile-verified
//
#include <hip/hip_runtime.h>

typedef __attribute__((ext_vector_type(2))) float v2f;
typedef __attribute__((ext_vector_type(8))) float v8f;

#define BB 256
#define NN 1152
#define II 8
#define CC 10
#define OO 16
#define WAVES 8
#define THREADS 256
#define NPW (NN / WAVES)   // 144 n-values per wave

// Build the priors fragment P (16 b x 16 o, f32 C/D layout) for one route node n:
//   P = A(16x8) x B(8x16) via two chained V_WMMA_F32_16X16X4_F32.
// A layout (ISA 16x4 f32): lane<16 -> K=0,1 in v[0],v[1]; lane>=16 -> K=2,3.
// xp must already point at x[b0+lo, n, 2*hi]; wp at W[c, n, 0, lo].
__device__ __forceinline__ v8f priors_frag(const float* __restrict__ xp,
                                           const float* __restrict__ wp,
                                           int hi) {
  v2f a0 = *(const v2f*)(xp);        // i = 2*hi + {0,1}
  v2f a1 = *(const v2f*)(xp + 4);    // i = 4 + 2*hi + {0,1}
  v2f bf0, bf1;                      // B: lane holds column o, rows K=2*hi, 2*hi+1
  bf0.x = wp[(2 * hi) * OO];
  bf0.y = wp[(2 * hi + 1) * OO];
  bf1.x = wp[(4 + 2 * hi) * OO];
  bf1.y = wp[(5 + 2 * hi) * OO];
  v8f p = {0.f, 0.f, 0.f, 0.f, 0.f, 0.f, 0.f, 0.f};
  p = __builtin_amdgcn_wmma_f32_16x16x4_f32(false, a0, false, bf0, (short)0, p, false, false);
  p = __builtin_amdgcn_wmma_f32_16x16x4_f32(false, a1, false, bf1, (short)0, p, false, false);
  return p;
}

__launch_bounds__(THREADS, 1)
__global__ void capsule_route_kernel(const float* __restrict__ x,
                                     const float* __restrict__ w,
                                     float* __restrict__ out) {
  __shared__ float lg[NN * 16];          // logits [n][b]  (72 KB)
  __shared__ float spart[WAVES][256];    // per-wave partial s [b*16+o]
  __shared__ float s_sh[256];            // s [b][o]
  __shared__ float out_sh[256];          // squash(s) [b][o]
  __shared__ float maxv[16];             // softmax max per b
  __shared__ float invv[16];             // 1/sum(exp) per b
  __shared__ float red[THREADS];         // reduction scratch

  const int t    = threadIdx.x;
  const int wave = t >> 5;
  const int lane = t & 31;
  const int lo   = lane & 15;            // column index (o for B/C frags, b-row for A)
  const int hi   = lane >> 4;            // half-wave select
  const int c    = blockIdx.y;
  const int b0   = blockIdx.x * 16;

  const float* wc = w + (size_t)c * NN * II * OO;
  const float* xb = x + ((size_t)(b0 + lo) * NN) * II + 2 * hi;

  // zero logits
  for (int k = t; k < NN * 16; k += THREADS) lg[k] = 0.0f;
  __syncthreads();

  const int bred = t & 15;               // b handled in n-reductions
  const int part = t >> 4;               // 16 partitions of n

  for (int iter = 0; iter < 3; ++iter) {
    // ---------- softmax stats over n (per b): max, 1/sum(exp) ----------
    float mx = -3.402823466e38f;
    for (int j = 0; j < NN / 16; ++j)
      mx = fmaxf(mx, lg[(part + 16 * j) * 16 + bred]);
    red[t] = mx;
    __syncthreads();
    if (t < 16) {
      float m2 = red[t];
      for (int q = 1; q < 16; ++q) m2 = fmaxf(m2, red[t + 16 * q]);
      maxv[t] = m2;
    }
    __syncthreads();
    float sm = 0.0f;
    {
      const float mb = maxv[bred];
      for (int j = 0; j < NN / 16; ++j)
        sm += __expf(lg[(part + 16 * j) * 16 + bred] - mb);
    }
    red[t] = sm;
    __syncthreads();
    if (t < 16) {
      float s2 = 0.0f;
      for (int q = 0; q < 16; ++q) s2 += red[t + 16 * q];
      invv[t] = 1.0f / s2;
    }
    __syncthreads();

    // ---------- pass A: s[b][o] = inv[b] * sum_n exp(lg-max) * P[n,b,o] ----------
    float mloc[8];
#pragma unroll
    for (int r = 0; r < 8; ++r) mloc[r] = maxv[r + 8 * hi];
    float acc[8] = {0.f, 0.f, 0.f, 0.f, 0.f, 0.f, 0.f, 0.f};
    for (int nn2 = 0; nn2 < NPW; ++nn2) {
      const int n = wave + nn2 * WAVES;
      v8f p = priors_frag(xb + (size_t)n * II, wc + (size_t)n * II * OO + lo, hi);
#pragma unroll
      for (int r = 0; r < 8; ++r) {
        float e = __expf(lg[n * 16 + r + 8 * hi] - mloc[r]);
        acc[r] += e * p[r];
      }
    }
#pragma unroll
    for (int r = 0; r < 8; ++r) spart[wave][(r + 8 * hi) * 16 + lo] = acc[r];
    __syncthreads();
    {
      float ss = 0.0f;
      for (int wv = 0; wv < WAVES; ++wv) ss += spart[wv][t];
      s_sh[t] = ss * invv[t >> 4];
    }
    __syncthreads();

    // ---------- squash ----------
    if (t < 16) {
      float sq = 0.0f;
      for (int o = 0; o < OO; ++o) { float v = s_sh[t * 16 + o]; sq += v * v; }
      const float scale = (sq / (1.0f + sq)) * rsqrtf(sq);
      for (int o = 0; o < OO; ++o) out_sh[t * 16 + o] = s_sh[t * 16 + o] * scale;
    }
    __syncthreads();

    if (iter == 2) break;

    // ---------- pass B: lg[n][b] += sum_o P[n,b,o] * out[b,o] ----------
    float ov[8];
#pragma unroll
    for (int r = 0; r < 8; ++r) ov[r] = out_sh[(r + 8 * hi) * 16 + lo];
    for (int nn2 = 0; nn2 < NPW; ++nn2) {
      const int n = wave + nn2 * WAVES;
      v8f p = priors_frag(xb + (size_t)n * II, wc + (size_t)n * II * OO + lo, hi);
#pragma unroll
      for (int r = 0; r < 8; ++r) {
        float tv = p[r] * ov[r];
        // sum over the 16 lanes holding o=0..15 for this b (xor masks stay in-group)
        tv += __shfl_xor(tv, 1, 32);
        tv += __shfl_xor(tv, 2, 32);
        tv += __shfl_xor(tv, 4, 32);
        tv += __shfl_xor(tv, 8, 32);
        if (lo == 0) lg[n * 16 + r + 8 * hi] += tv;
      }
    }
    __syncthreads();
  }

  // output [C,B,1,O]: thread t writes (b_local=t>>4, o=t&15)
  out[((size_t)c * BB + b0 + (t >> 4)) * OO + (t & 15)] = out_sh[t];
}

extern "C" void kernel_launch(void* const* d_in, const int* in_sizes, int n_in,
                              void* d_out, int out_size, void* d_ws, size_t ws_size,
                              hipStream_t stream) {
  (void)in_sizes; (void)n_in; (void)out_size; (void)d_ws; (void)ws_size;
  const float* x = (const float*)d_in[0];            // [256,1152,8]
  const float* w = (const float*)d_in[1];            // [10,1152,8,16]
  float* out = (float*)d_out;                        // [10,256,1,16]
  dim3 grid(BB / 16, CC);
  capsule_route_kernel<<<grid, dim3(THREADS), 0, stream>>>(x, w, out);
}